// HungarianMatcher_25331717112271
// MI455X (gfx1250) — compile-verified
//
#include <hip/hip_runtime.h>
#include <math.h>

typedef __attribute__((ext_vector_type(2))) float v2f;
typedef __attribute__((ext_vector_type(8))) float v8f;

#define BS   16
#define NQ   900
#define NC   151
#define NT   100
#define NQT  (BS*NQ)     // 14400 query rows
#define TT   (BS*NT)     // 1600 targets
#define NCP  152         // K padded to multiple of 4 for WMMA
#define LSTR 154         // LDS row stride (floats): r*154 mod 64 distinct for 16 rows
#define TPB  128         // 4 wave32 waves
#define TBLK 320         // targets per block: 4 waves * 5 tiles * 16

__global__ __launch_bounds__(TPB) void detr_cost_kernel(
    const float* __restrict__ logits,   // [NQT, NC]
    const float* __restrict__ pboxes,   // [NQT, 4]  cxcywh
    const int*   __restrict__ tlabels,  // [TT]
    const float* __restrict__ tboxes,   // [TT, 4]   cxcywh
    float* __restrict__ out)            // [NQT, TT]
{
    __shared__ float  probsS[16 * LSTR];
    __shared__ float4 predS[16];
    __shared__ float4 tgtS[TBLK];
    __shared__ int    idsS[TBLK];

    const int tid    = threadIdx.x;
    const int lane   = tid & 31;
    const int w      = tid >> 5;               // wave id 0..3
    const int qbase  = blockIdx.y * 16;
    const int tbase0 = blockIdx.x * TBLK;

    // ---------------- softmax of 16 query rows -> LDS ----------------
    {
        const int rloc = (w << 2) + (lane >> 3);   // 0..15 local row
        const int row  = qbase + rloc;
        const int c0   = lane & 7;                 // 8 lanes per row
        float x[19];
        float mx = -3.4e38f;
#pragma unroll
        for (int i = 0; i < 19; i++) {
            int c = c0 + (i << 3);
            float v = (c < NC) ? logits[row * NC + c] : -3.4e38f;
            x[i] = v;
            mx = fmaxf(mx, v);
        }
        mx = fmaxf(mx, __shfl_xor(mx, 1, 32));
        mx = fmaxf(mx, __shfl_xor(mx, 2, 32));
        mx = fmaxf(mx, __shfl_xor(mx, 4, 32));
        float s = 0.f;
#pragma unroll
        for (int i = 0; i < 19; i++) {
            int c = c0 + (i << 3);
            float e = (c < NC) ? __expf(x[i] - mx) : 0.f;
            x[i] = e;
            s += e;
        }
        s += __shfl_xor(s, 1, 32);
        s += __shfl_xor(s, 2, 32);
        s += __shfl_xor(s, 4, 32);
        const float inv = 1.0f / s;
#pragma unroll
        for (int i = 0; i < 19; i++) {
            int c = c0 + (i << 3);
            if (c < NCP) probsS[rloc * LSTR + c] = (c < NC) ? x[i] * inv : 0.f;
        }
    }

    // ---------------- stage boxes / labels ----------------
    if (tid < 16) predS[tid] = ((const float4*)pboxes)[qbase + tid];
    for (int t = tid; t < TBLK; t += TPB) {
        tgtS[t] = ((const float4*)tboxes)[tbase0 + t];
        idsS[t] = tlabels[tbase0 + t];
    }
    __syncthreads();

    const int h = lane >> 4;   // half of wave -> K phase / row+8 select
    const int r = lane & 15;

    // Hoist per-lane pred rows (m = v + 8h), reused across all 5 t-tiles.
    float4 praw[8];
    float px0[8], py0[8], px1[8], py1[8], pa[8];
#pragma unroll
    for (int v = 0; v < 8; v++) {
        float4 pb = predS[v + (h << 3)];
        praw[v] = pb;
        px0[v] = pb.x - 0.5f * pb.z;  px1[v] = pb.x + 0.5f * pb.z;
        py0[v] = pb.y - 0.5f * pb.w;  py1[v] = pb.y + 0.5f * pb.w;
        pa[v]  = pb.z * pb.w;
    }

    for (int j = 0; j < 5; j++) {
        const int tl = j * 64 + (w << 4);       // local t-tile base
        const int id = idsS[tl + r];            // class id for this lane's column

        // ---- class prob via f32 WMMA: D = P(16xK) x OneHot(Kx16), K = 152 ----
        // A 16x4 f32 layout: lane r holds row M=r; VGPR j -> K = 4*ks + 2h + j
        // B  4x16 f32 layout: lane r holds col N=r; VGPR j -> K = 4*ks + 2h + j
        v8f acc = {};
        int k0 = (h << 1);
        const float* aptr = &probsS[r * LSTR + (h << 1)];
#pragma unroll
        for (int ks = 0; ks < 38; ks++) {
            v2f a = *(const v2f*)(aptr + (ks << 2));
            v2f b;
            b.x = (id == k0    ) ? 1.f : 0.f;
            b.y = (id == k0 + 1) ? 1.f : 0.f;
            acc = __builtin_amdgcn_wmma_f32_16x16x4_f32(
                false, a, false, b, (short)0, acc, false, false);
            k0 += 4;
        }

        // ---- L1 + GIoU epilogue ----
        float4 tb = tgtS[tl + r];
        float tx0 = tb.x - 0.5f * tb.z, tx1 = tb.x + 0.5f * tb.z;
        float ty0 = tb.y - 0.5f * tb.w, ty1 = tb.y + 0.5f * tb.w;
        float ta  = tb.z * tb.w;
        const int tglob = tbase0 + tl + r;
#pragma unroll
        for (int v = 0; v < 8; v++) {
            float4 pb = praw[v];
            float l1 = fabsf(pb.x - tb.x) + fabsf(pb.y - tb.y) +
                       fabsf(pb.z - tb.z) + fabsf(pb.w - tb.w);
            float ix0 = fmaxf(px0[v], tx0), iy0 = fmaxf(py0[v], ty0);
            float ix1 = fminf(px1[v], tx1), iy1 = fminf(py1[v], ty1);
            float iw  = fmaxf(ix1 - ix0, 0.f), ih = fmaxf(iy1 - iy0, 0.f);
            float inter = iw * ih;
            float uni   = pa[v] + ta - inter;
            float ex0 = fminf(px0[v], tx0), ey0 = fminf(py0[v], ty0);
            float ex1 = fmaxf(px1[v], tx1), ey1 = fmaxf(py1[v], ty1);
            float ea  = (ex1 - ex0) * (ey1 - ey0);
            // giou = inter/uni - (ea-uni)/ea = inter/uni + uni/ea - 1
            float giou = inter * __builtin_amdgcn_rcpf(uni) +
                         uni   * __builtin_amdgcn_rcpf(ea) - 1.f;
            // C = 5*l1 + 1*(-prob) + 2*(-giou)
            float c = 5.f * l1 - acc[v] - 2.f * giou;
            int m = v + (h << 3);
            __builtin_nontemporal_store(
                c, &out[(size_t)(qbase + m) * TT + tglob]);
        }
    }
}

extern "C" void kernel_launch(void* const* d_in, const int* in_sizes, int n_in,
                              void* d_out, int out_size, void* d_ws, size_t ws_size,
                              hipStream_t stream) {
    (void)in_sizes; (void)n_in; (void)out_size; (void)d_ws; (void)ws_size;
    const float* logits  = (const float*)d_in[0];
    const float* pboxes  = (const float*)d_in[1];
    const int*   tlabels = (const int*)d_in[2];
    const float* tboxes  = (const float*)d_in[3];
    float* out = (float*)d_out;

    dim3 grid(TT / TBLK /* 5 */, NQT / 16 /* 900 */);
    detr_cost_kernel<<<grid, TPB, 0, stream>>>(logits, pboxes, tlabels, tboxes, out);
}